// NoiseMixModule_33586644255147
// MI455X (gfx1250) — compile-verified
//
#include <hip/hip_runtime.h>
#include <math.h>

// NoiseMix: out = mask * A + (1-mask) * B, mask from per-row rank of u vs
// num_zero = floor(7*(1-lam)).  Pure HBM-bound (~487 MB traffic -> ~21us at
// 23.3 TB/s).  Strategy: stage the stride-7 AoS rows through LDS using CDNA5
// async global<->LDS b128 copies (coalesced, 16B aligned), do the 7x7 stable
// rank network in VALU, write results through LDS with async stores.

#define NOISE_DIM 7
#define ROWS_PER_BLOCK 256
#define CHUNK_BYTES (ROWS_PER_BLOCK * NOISE_DIM * 4)   // 7168 B per array per block
#define LDS_FLOATS (3 * ROWS_PER_BLOCK * NOISE_DIM)    // 5376 floats = 21504 B

__device__ __forceinline__ void async_g2l_b128(unsigned lds_addr, unsigned goff,
                                               const void* sbase) {
    // GLOBAL_LOAD_ASYNC_TO_LDS_B128 (GVS mode): 16 bytes from global at saddr plus vaddr to LDS at vdst
    asm volatile("global_load_async_to_lds_b128 %0, %1, %2"
                 :: "v"(lds_addr), "v"(goff), "s"(sbase) : "memory");
}

__device__ __forceinline__ void async_l2g_b128(unsigned goff, unsigned lds_addr,
                                               void* sbase) {
    // GLOBAL_STORE_ASYNC_FROM_LDS_B128 (GVS mode): 16 bytes from LDS at vsrc to global at saddr plus vaddr
    asm volatile("global_store_async_from_lds_b128 %0, %1, %2"
                 :: "v"(goff), "v"(lds_addr), "s"(sbase) : "memory");
}

__device__ __forceinline__ void wait_async0() {
    asm volatile("s_wait_asynccnt 0" ::: "memory");
}

__global__ __launch_bounds__(ROWS_PER_BLOCK)
void noisemix_kernel(const float* __restrict__ nA, const float* __restrict__ nB,
                     const float* __restrict__ lam, const float* __restrict__ u,
                     float* __restrict__ out, long long n_rows) {
    __shared__ __align__(16) float lds[LDS_FLOATS];

    const int tid = threadIdx.x;
    const long long r0 = (long long)blockIdx.x * ROWS_PER_BLOCK;

    if (r0 + ROWS_PER_BLOCK <= n_rows) {
        // ---------------- fast path: full block of 256 rows ----------------
        const float* srcA = nA + r0 * NOISE_DIM;
        const float* srcB = nB + r0 * NOISE_DIM;
        const float* srcU = u  + r0 * NOISE_DIM;
        float*       dst  = out + r0 * NOISE_DIM;

        // low 32 bits of the flat shared address == LDS byte offset
        const unsigned ldsbase = (unsigned)(size_t)(const void*)&lds[0];
        const unsigned off     = (unsigned)tid * 16u;

        // stage A, B, U: 448 x b128 per array (256 + 192 per round)
        async_g2l_b128(ldsbase + off, off, srcA);
        async_g2l_b128(ldsbase + CHUNK_BYTES + off, off, srcB);
        async_g2l_b128(ldsbase + 2u * CHUNK_BYTES + off, off, srcU);
        if (tid < 192) {
            const unsigned off2 = off + 4096u;
            async_g2l_b128(ldsbase + off2, off2, srcA);
            async_g2l_b128(ldsbase + CHUNK_BYTES + off2, off2, srcB);
            async_g2l_b128(ldsbase + 2u * CHUNK_BYTES + off2, off2, srcU);
        }
        wait_async0();
        __syncthreads();

        // coalesced b32 load of lam
        const float lamv = lam[r0 + tid];
        const int num_zero = (int)floorf((float)NOISE_DIM * (1.0f - lamv));

        // read own row of u (stride-7 dwords -> bank-conflict free, gcd(7,64)=1)
        float uu[NOISE_DIM];
#pragma unroll
        for (int d = 0; d < NOISE_DIM; ++d)
            uu[d] = lds[2 * ROWS_PER_BLOCK * NOISE_DIM + tid * NOISE_DIM + d];

        float res[NOISE_DIM];
#pragma unroll
        for (int d = 0; d < NOISE_DIM; ++d) {
            // stable rank: # of elements strictly before uu[d] in a stable sort
            int rank = 0;
#pragma unroll
            for (int e = 0; e < NOISE_DIM; ++e) {
                const bool before =
                    (uu[e] < uu[d]) || ((uu[e] == uu[d]) && (e < d));
                rank += before ? 1 : 0;
            }
            const float a = lds[tid * NOISE_DIM + d];
            const float b = lds[ROWS_PER_BLOCK * NOISE_DIM + tid * NOISE_DIM + d];
            res[d] = (rank >= num_zero) ? a : b;
        }

        // write result into the (dead) A region of LDS, own row only
#pragma unroll
        for (int d = 0; d < NOISE_DIM; ++d)
            lds[tid * NOISE_DIM + d] = res[d];
        __syncthreads();

        // async b128 stores LDS -> global (coalesced output)
        async_l2g_b128(off, ldsbase + off, dst);
        if (tid < 192) {
            const unsigned off2 = off + 4096u;
            async_l2g_b128(off2, ldsbase + off2, dst);
        }
        wait_async0();
    } else {
        // ---------------- tail path: guarded scalar (block-uniform branch) ----
        const long long row = r0 + tid;
        if (row < n_rows) {
            const float lamv = lam[row];
            const int num_zero = (int)floorf((float)NOISE_DIM * (1.0f - lamv));
            float uu[NOISE_DIM];
#pragma unroll
            for (int d = 0; d < NOISE_DIM; ++d)
                uu[d] = u[row * NOISE_DIM + d];
#pragma unroll
            for (int d = 0; d < NOISE_DIM; ++d) {
                int rank = 0;
#pragma unroll
                for (int e = 0; e < NOISE_DIM; ++e) {
                    const bool before =
                        (uu[e] < uu[d]) || ((uu[e] == uu[d]) && (e < d));
                    rank += before ? 1 : 0;
                }
                const float a = nA[row * NOISE_DIM + d];
                const float b = nB[row * NOISE_DIM + d];
                out[row * NOISE_DIM + d] = (rank >= num_zero) ? a : b;
            }
        }
    }
}

extern "C" void kernel_launch(void* const* d_in, const int* in_sizes, int n_in,
                              void* d_out, int out_size, void* d_ws, size_t ws_size,
                              hipStream_t stream) {
    const float* nA  = (const float*)d_in[0];
    const float* nB  = (const float*)d_in[1];
    const float* lam = (const float*)d_in[2];
    const float* u   = (const float*)d_in[3];
    float* out = (float*)d_out;

    const long long n_rows = (long long)in_sizes[2];  // lam has one entry per row
    const int blocks = (int)((n_rows + ROWS_PER_BLOCK - 1) / ROWS_PER_BLOCK);

    noisemix_kernel<<<blocks, ROWS_PER_BLOCK, 0, stream>>>(nA, nB, lam, u, out,
                                                           n_rows);
}